// NeuroSparseTransformer_11458972746042
// MI455X (gfx1250) — compile-verified
//
#include <hip/hip_runtime.h>
#include <hip/hip_bf16.h>
#include <stdint.h>

#define SEQ   4096
#define DIM   256
#define NEXP  4
#define NLAY  4
#define VOC   32000
#define KGATE 2201   // int(4096 * sigmoid(0.15))

// head GEMM blocking
#define TM 64        // block tile M (tokens)
#define TN 256       // block tile N (vocab)
#define KC 32        // K chunk staged in LDS (double-buffered)
#define PAD 4        // LDS row padding (floats) -> bank-conflict free

typedef __attribute__((ext_vector_type(2))) float v2f;
typedef __attribute__((ext_vector_type(8))) float v8f;

// ---------------- embedding gather ----------------
__global__ void embed_kernel(const int* __restrict__ tokens,
                             const float* __restrict__ emb,
                             float* __restrict__ x) {
    int s = blockIdx.x, d = threadIdx.x;
    x[s * DIM + d] = emb[(long)tokens[s] * DIM + d];
}

// ---------------- AstroNorm column stats (mean/unbiased var over S per feature) ----------------
__global__ void stats_kernel(const float* __restrict__ x,
                             const float* __restrict__ buffers,
                             const float* __restrict__ decays, int l,
                             float* __restrict__ mu, float* __restrict__ rsig,
                             float* __restrict__ nbabs) {
    int d = blockIdx.x, t = threadIdx.x;
    float sum = 0.f, sq = 0.f;
    for (int s = t; s < SEQ; s += blockDim.x) {
        float v = x[s * DIM + d];
        sum += v; sq += v * v;
    }
    __shared__ float ls[256], lq[256];
    ls[t] = sum; lq[t] = sq; __syncthreads();
    for (int o = 128; o > 0; o >>= 1) {
        if (t < o) { ls[t] += ls[t + o]; lq[t] += lq[t + o]; }
        __syncthreads();
    }
    if (t == 0) {
        float m   = ls[0] / (float)SEQ;
        float var = (lq[0] - (float)SEQ * m * m) / (float)(SEQ - 1);  // ddof=1
        mu[d]   = m;
        rsig[d] = rsqrtf(var + 1e-6f);
        float dec = decays[l];
        nbabs[d] = fabsf(dec * buffers[l * DIM + d] + (1.f - dec) * m);
    }
}

// ---------------- AstroNorm apply + gate score (fused) ----------------
__global__ void astro_gate_kernel(float* __restrict__ x,
                                  const float* __restrict__ mu,
                                  const float* __restrict__ rsig,
                                  const float* __restrict__ nbabs,
                                  const float* __restrict__ gate_w, int l,
                                  float* __restrict__ scores) {
    int s = blockIdx.x, t = threadIdx.x;
    float m  = mu[t];
    float xv = x[s * DIM + t];
    float v  = (fabsf(xv - m) > nbabs[t] ? m : xv) * rsig[t];
    x[s * DIM + t] = v;
    __shared__ float red[256];
    red[t] = v * gate_w[l * DIM + t];
    __syncthreads();
    for (int o = 128; o > 0; o >>= 1) {
        if (t < o) red[t] += red[t + o];
        __syncthreads();
    }
    if (t == 0) scores[s] = red[0];
}

// ---------------- k-th largest of 4096 scores: bitonic sort in LDS, 1 workgroup ----------------
__global__ void topk_thr_kernel(const float* __restrict__ scores, float* __restrict__ thr) {
    __shared__ float sh[SEQ];
    int t = threadIdx.x;  // 1024 threads
    for (int i = t; i < SEQ; i += 1024) sh[i] = scores[i];
    __syncthreads();
    for (int k = 2; k <= SEQ; k <<= 1) {
        for (int j = k >> 1; j > 0; j >>= 1) {
            for (int i = t; i < SEQ; i += 1024) {
                int ix = i ^ j;
                if (ix > i) {
                    bool up = ((i & k) == 0);
                    float a = sh[i], b = sh[ix];
                    if (up == (a > b)) { sh[i] = b; sh[ix] = a; }
                }
            }
            __syncthreads();
        }
    }
    if (t == 0) thr[0] = sh[SEQ - KGATE];  // ascending sort -> KGATE-th largest
}

// ---------------- gate mask + expert routing (top-3-of-4 => exclude argmin) ----------------
__global__ void gate_route_kernel(float* __restrict__ x,
                                  const float* __restrict__ scores,
                                  const float* __restrict__ thr,
                                  const float* __restrict__ gW,
                                  const float* __restrict__ gb, int l,
                                  float* __restrict__ emask) {
    int s = blockIdx.x, t = threadIdx.x;
    float keep = scores[s] > thr[0] ? 1.f : 0.f;   // strict >, ties at thr dropped
    float xv = x[s * DIM + t] * keep;
    x[s * DIM + t] = xv;
    __shared__ float red[256];
    __shared__ float gs[NEXP];
    for (int e = 0; e < NEXP; e++) {
        red[t] = xv * gW[(l * NEXP + e) * DIM + t];
        __syncthreads();
        for (int o = 128; o > 0; o >>= 1) {
            if (t < o) red[t] += red[t + o];
            __syncthreads();
        }
        if (t == 0) gs[e] = red[0] + gb[l * NEXP + e];
        __syncthreads();
    }
    if (t == 0) {
        int ex = 0; float mv = gs[0];
        for (int e = 1; e < NEXP; e++) if (gs[e] <= mv) { mv = gs[e]; ex = e; }
        for (int e = 0; e < NEXP; e++) emask[s * NEXP + e] = (e == ex) ? 0.f : 1.f;
    }
}

// helper: issue one 16B async global->LDS copy (gfx1250 ASYNCcnt path)
__device__ __forceinline__ void async_copy16(unsigned lds_off, const float* gptr) {
    unsigned long long ga = (unsigned long long)(uintptr_t)gptr;
    asm volatile("global_load_async_to_lds_b128 %0, %1, off"
                 :: "v"(lds_off), "v"(ga) : "memory");
}
__device__ __forceinline__ void async_wait0() {
    asm volatile("s_wait_asynccnt 0" ::: "memory");
}

// ---------------- per-layer expert GEMM, LDS-tiled + async-pipelined ----------------
// grid: SEQ/TM = 64 blocks; block 256 threads = 8 waves as 2(M) x 4(N)
// block tile: 64(M) x 256(N = full DIM); A staged once (full K), B double-buffered
// over 32 stages = 4 experts x 8 K-chunks. Masked per-expert accumulation.
__global__ void __launch_bounds__(256) expert_gemm_kernel(
        const float* __restrict__ X,
        const float* __restrict__ W,    // [E][DIM][DIM] for this layer
        const float* __restrict__ bvec, // [E][DIM]
        const float* __restrict__ emask,// [SEQ][E]
        float* __restrict__ Y) {
    __shared__ float sA[TM][DIM + PAD];        // 64 x 260 floats (~67 KB)
    __shared__ float sB[2][DIM][KC + PAD];     // 2 x 256 x 36 floats (~74 KB)
    int tid  = threadIdx.x;
    int wave = tid >> 5, lane = tid & 31;
    int wm = wave >> 2, wn = wave & 3;
    int row = lane & 15, half = lane >> 4;
    int gm0 = blockIdx.x * TM;

    // stage A once: 64 rows x 256 cols, 4 threads/row, 64 floats each
    {
        int r = tid >> 2, c0 = (tid & 3) * 64;
        const float* src = X + (size_t)(gm0 + r) * DIM + c0;
        unsigned la = (unsigned)(uintptr_t)&sA[r][c0];
#pragma unroll
        for (int i = 0; i < 16; i++) async_copy16(la + i * 16, src + i * 4);
    }
    // stage B stage 0 (e=0, kc=0): one DIM-row per thread, KC floats
    {
        const float* src = W + (size_t)tid * DIM;
        unsigned lb = (unsigned)(uintptr_t)&sB[0][tid][0];
#pragma unroll
        for (int i = 0; i < KC / 4; i++) async_copy16(lb + i * 16, src + i * 4);
    }
    async_wait0();
    __syncthreads();

    float outacc[2][4][8];
#pragma unroll
    for (int i = 0; i < 2; i++)
#pragma unroll
        for (int j = 0; j < 4; j++)
#pragma unroll
            for (int q = 0; q < 8; q++) outacc[i][j][q] = 0.f;

    v8f c[2][4];
#pragma unroll
    for (int i = 0; i < 2; i++)
#pragma unroll
        for (int j = 0; j < 4; j++) c[i][j] = (v8f){};

    const int NSTAGE = NEXP * (DIM / KC);   // 32
    for (int s = 0; s < NSTAGE; s++) {
        int e  = s >> 3;
        int kc = (s & 7) * KC;
        // prefetch next stage's B chunk into the other buffer
        if (s + 1 < NSTAGE) {
            int en  = (s + 1) >> 3;
            int kcn = ((s + 1) & 7) * KC;
            const float* src = W + (size_t)en * DIM * DIM + (size_t)tid * DIM + kcn;
            unsigned lb = (unsigned)(uintptr_t)&sB[(s + 1) & 1][tid][0];
#pragma unroll
            for (int i = 0; i < KC / 4; i++) async_copy16(lb + i * 16, src + i * 4);
        }
        // compute on current stage
        for (int k = 0; k < KC; k += 4) {
            v2f a[2];
#pragma unroll
            for (int i = 0; i < 2; i++) {
                a[i].x = sA[wm * 32 + i * 16 + row][kc + k + half * 2];
                a[i].y = sA[wm * 32 + i * 16 + row][kc + k + half * 2 + 1];
            }
#pragma unroll
            for (int j = 0; j < 4; j++) {
                v2f b;
                b.x = sB[s & 1][wn * 64 + j * 16 + row][k + half * 2];
                b.y = sB[s & 1][wn * 64 + j * 16 + row][k + half * 2 + 1];
#pragma unroll
                for (int i = 0; i < 2; i++)
                    c[i][j] = __builtin_amdgcn_wmma_f32_16x16x4_f32(
                        false, a[i], false, b, (short)0, c[i][j], false, false);
            }
        }
        // expert finished -> masked accumulate, reset c
        if ((s & 7) == 7) {
#pragma unroll
            for (int j = 0; j < 4; j++) {
                int n = wn * 64 + j * 16 + row;
                float be = bvec[e * DIM + n];
#pragma unroll
                for (int i = 0; i < 2; i++) {
#pragma unroll
                    for (int q = 0; q < 8; q++) {
                        int m = gm0 + wm * 32 + i * 16 + q + half * 8;
                        float mk = emask[(size_t)m * NEXP + e];
                        outacc[i][j][q] += mk * (c[i][j][q] + be);
                    }
                }
            }
#pragma unroll
            for (int i = 0; i < 2; i++)
#pragma unroll
                for (int j = 0; j < 4; j++) c[i][j] = (v8f){};
        }
        if (s + 1 < NSTAGE) async_wait0();
        __syncthreads();
    }

#pragma unroll
    for (int j = 0; j < 4; j++) {
        int n = wn * 64 + j * 16 + row;
#pragma unroll
        for (int i = 0; i < 2; i++) {
#pragma unroll
            for (int q = 0; q < 8; q++) {
                int m = gm0 + wm * 32 + i * 16 + q + half * 8;
                Y[(size_t)m * DIM + n] = outacc[i][j][q];
            }
        }
    }
}

// ---------------- final LayerNorm (biased var, eps=1e-5) ----------------
__global__ void ln_kernel(const float* __restrict__ x,
                          const float* __restrict__ g,
                          const float* __restrict__ b,
                          float* __restrict__ xn) {
    int s = blockIdx.x, t = threadIdx.x;
    float v = x[s * DIM + t];
    __shared__ float rs[256], rq[256];
    rs[t] = v; rq[t] = v * v; __syncthreads();
    for (int o = 128; o > 0; o >>= 1) {
        if (t < o) { rs[t] += rs[t + o]; rq[t] += rq[t + o]; }
        __syncthreads();
    }
    float m   = rs[0] / (float)DIM;
    float var = rq[0] / (float)DIM - m * m;
    xn[s * DIM + t] = (v - m) * rsqrtf(var + 1e-5f) * g[t] + b[t];
}

// ---------------- vocab head GEMM: LDS-tiled, double-buffered async pipeline ----------------
// grid: (SEQ/TM, VOC/TN) = (64, 125); block 256 threads = 8 waves as 2(M) x 4(N)
// wave tile: 32(M) x 64(N) -> 2x4 v8f accumulators; KC=32 chunks, ping-pong LDS
__global__ void __launch_bounds__(256) head_kernel(const float* __restrict__ Xn,
                                                   const float* __restrict__ hW,  // [VOC][DIM]
                                                   const float* __restrict__ hb,  // [VOC]
                                                   float* __restrict__ out) {
    __shared__ float sA[2][TM][KC + PAD];   // 2 x 64 x 36
    __shared__ float sB[2][TN][KC + PAD];   // 2 x 256 x 36
    int tid  = threadIdx.x;
    int wave = tid >> 5, lane = tid & 31;
    int wm = wave >> 2, wn = wave & 3;
    int row = lane & 15, half = lane >> 4;
    int gm0 = blockIdx.x * TM;
    int gn0 = blockIdx.y * TN;

    v8f acc[2][4];
#pragma unroll
    for (int i = 0; i < 2; i++)
#pragma unroll
        for (int j = 0; j < 4; j++) acc[i][j] = (v8f){};

    int ar  = tid >> 2;            // A row 0..63 (4 threads/row)
    int asg = (tid & 3) * (KC / 4) * 2;  // 8 floats per thread: cols (tid&3)*16

    // issue stage 0
    {
        const float* srcA = Xn + (size_t)(gm0 + ar) * DIM + asg;
        unsigned la = (unsigned)(uintptr_t)&sA[0][ar][asg];
#pragma unroll
        for (int i = 0; i < 2; i++) async_copy16(la + i * 16, srcA + i * 4);
        const float* srcB = hW + (size_t)(gn0 + tid) * DIM;
        unsigned lb = (unsigned)(uintptr_t)&sB[0][tid][0];
#pragma unroll
        for (int i = 0; i < KC / 4; i++) async_copy16(lb + i * 16, srcB + i * 4);
    }
    async_wait0();
    __syncthreads();

    const int NC = DIM / KC;   // 8 chunks
    for (int cchunk = 0; cchunk < NC; cchunk++) {
        // prefetch next chunk into the other buffer (overlaps with WMMA below)
        if (cchunk + 1 < NC) {
            int kcn = (cchunk + 1) * KC;
            const float* srcA = Xn + (size_t)(gm0 + ar) * DIM + kcn + asg;
            unsigned la = (unsigned)(uintptr_t)&sA[(cchunk + 1) & 1][ar][asg];
#pragma unroll
            for (int i = 0; i < 2; i++) async_copy16(la + i * 16, srcA + i * 4);
            const float* srcB = hW + (size_t)(gn0 + tid) * DIM + kcn;
            unsigned lb = (unsigned)(uintptr_t)&sB[(cchunk + 1) & 1][tid][0];
#pragma unroll
            for (int i = 0; i < KC / 4; i++) async_copy16(lb + i * 16, srcB + i * 4);
        }
        int buf = cchunk & 1;
        for (int k = 0; k < KC; k += 4) {
            v2f a[2];
#pragma unroll
            for (int i = 0; i < 2; i++) {
                a[i].x = sA[buf][wm * 32 + i * 16 + row][k + half * 2];
                a[i].y = sA[buf][wm * 32 + i * 16 + row][k + half * 2 + 1];
            }
#pragma unroll
            for (int j = 0; j < 4; j++) {
                v2f b;
                b.x = sB[buf][wn * 64 + j * 16 + row][k + half * 2];
                b.y = sB[buf][wn * 64 + j * 16 + row][k + half * 2 + 1];
#pragma unroll
                for (int i = 0; i < 2; i++)
                    acc[i][j] = __builtin_amdgcn_wmma_f32_16x16x4_f32(
                        false, a[i], false, b, (short)0, acc[i][j], false, false);
            }
        }
        if (cchunk + 1 < NC) async_wait0();
        __syncthreads();
    }

    // epilogue: +bias, store (lanes 0-15 give 64B-coalesced rows)
#pragma unroll
    for (int j = 0; j < 4; j++) {
        int n = gn0 + wn * 64 + j * 16 + row;
        float bb = hb[n];
#pragma unroll
        for (int i = 0; i < 2; i++) {
#pragma unroll
            for (int q = 0; q < 8; q++) {
                int m = gm0 + wm * 32 + i * 16 + q + half * 8;
                out[(size_t)m * VOC + n] = acc[i][j][q] + bb;
            }
        }
    }
}

extern "C" void kernel_launch(void* const* d_in, const int* in_sizes, int n_in,
                              void* d_out, int out_size, void* d_ws, size_t ws_size,
                              hipStream_t stream) {
    const int*   tokens   = (const int*)  d_in[0];
    const float* emb      = (const float*)d_in[1];
    const float* buffers  = (const float*)d_in[2];
    const float* decays   = (const float*)d_in[3];
    const float* gate_w   = (const float*)d_in[4];
    const float* gating_W = (const float*)d_in[5];
    const float* gating_b = (const float*)d_in[6];
    const float* exp_W    = (const float*)d_in[7];
    const float* exp_b    = (const float*)d_in[8];
    const float* ln_g     = (const float*)d_in[9];
    const float* ln_b     = (const float*)d_in[10];
    const float* head_W   = (const float*)d_in[11];
    const float* head_b   = (const float*)d_in[12];
    float* out = (float*)d_out;

    float* ws     = (float*)d_ws;
    float* x0     = ws;                  // SEQ*DIM
    float* x1     = x0 + SEQ * DIM;      // SEQ*DIM
    float* mu     = x1 + SEQ * DIM;      // DIM
    float* rsig   = mu + DIM;            // DIM
    float* nbabs  = rsig + DIM;          // DIM
    float* scores = nbabs + DIM;         // SEQ
    float* thr    = scores + SEQ;        // 1 (+pad)
    float* emask  = thr + 16;            // SEQ*NEXP

    embed_kernel<<<SEQ, DIM, 0, stream>>>(tokens, emb, x0);

    float* xin = x0; float* xout = x1;
    for (int l = 0; l < NLAY; l++) {
        stats_kernel<<<DIM, 256, 0, stream>>>(xin, buffers, decays, l, mu, rsig, nbabs);
        astro_gate_kernel<<<SEQ, DIM, 0, stream>>>(xin, mu, rsig, nbabs, gate_w, l, scores);
        topk_thr_kernel<<<1, 1024, 0, stream>>>(scores, thr);
        gate_route_kernel<<<SEQ, DIM, 0, stream>>>(xin, scores, thr, gating_W, gating_b, l, emask);
        expert_gemm_kernel<<<SEQ / TM, 256, 0, stream>>>(
            xin, exp_W + (long)l * NEXP * DIM * DIM, exp_b + (long)l * NEXP * DIM, emask, xout);
        float* tmp = xin; xin = xout; xout = tmp;
    }

    ln_kernel<<<SEQ, DIM, 0, stream>>>(xin, ln_g, ln_b, xout);

    dim3 hgrid(SEQ / TM, VOC / TN);   // (64, 125)
    head_kernel<<<hgrid, 256, 0, stream>>>(xout, head_W, head_b, out);
}